// RRGCN_1460288881436
// MI455X (gfx1250) — compile-verified
//
#include <hip/hip_runtime.h>

// ---------------------------------------------------------------------------
// Fused LSTM-gate elementwise kernel (the RGCN convs are identity):
//   S      = X + H
//   sig    = sigmoid(S)          (I == F == O, computed once)
//   T      = tanh(S)
//   C_new  = sig * (C + T)
//   H_new  = sig * tanh(C_new)
// Outputs concatenated: d_out[0..n) = H_new, d_out[n..2n) = C_new.
//
// Pure streaming: 20 B/elem, ~1.02 GB total -> ~44 us floor at 23.3 TB/s.
// b128 non-temporal loads/stores, native v_tanh_f32 (3 trans/elem),
// flat one-float4-per-thread launch, 32-bit addressing.
// ---------------------------------------------------------------------------

typedef float v4f __attribute__((ext_vector_type(4)));

#if __has_builtin(__builtin_amdgcn_tanhf)
__device__ __forceinline__ float dev_tanh(float x) {
  return __builtin_amdgcn_tanhf(x);            // v_tanh_f32 (gfx1250 TRANS)
}
#elif __has_builtin(__builtin_amdgcn_tanh_f32)
__device__ __forceinline__ float dev_tanh(float x) {
  return __builtin_amdgcn_tanh_f32(x);
}
#else
__device__ __forceinline__ float dev_tanh(float x) {
  // tanh(x) = 1 - 2/(exp(2x)+1)   -> v_exp_f32 + v_rcp_f32
  float e = __expf(2.0f * x);
  return 1.0f - 2.0f * __builtin_amdgcn_rcpf(e + 1.0f);
}
#endif

// sigmoid(x) = 0.5*tanh(0.5x) + 0.5  (one trans op instead of exp+div)
__device__ __forceinline__ float dev_sigmoid(float x) {
  return fmaf(0.5f, dev_tanh(0.5f * x), 0.5f);
}

__device__ __forceinline__ void gate_update(float x, float h, float c,
                                            float& hn, float& cn) {
  float s   = x + h;
  float sig = dev_sigmoid(s);      // I == F == O
  float t   = dev_tanh(s);         // candidate
  float cne = sig * (c + t);       // F*C + I*T with F==I==sig
  cn = cne;
  hn = sig * dev_tanh(cne);        // O * tanh(C_new)
}

__global__ __launch_bounds__(256) void rrgcn_gates_v4(
    const v4f* __restrict__ X, const v4f* __restrict__ H,
    const v4f* __restrict__ C, v4f* __restrict__ Hout,
    v4f* __restrict__ Cout, unsigned nvec) {
  unsigned i = blockIdx.x * blockDim.x + threadIdx.x;  // 32-bit: max off 205 MB
  if (i >= nvec) return;
  // Single-touch streaming data: non-temporal b128 accesses, keep L2 clean.
  v4f x = __builtin_nontemporal_load(&X[i]);
  v4f h = __builtin_nontemporal_load(&H[i]);
  v4f c = __builtin_nontemporal_load(&C[i]);
  v4f hn, cn;
#pragma unroll
  for (int k = 0; k < 4; ++k) {
    float ho, co;
    gate_update(x[k], h[k], c[k], ho, co);
    hn[k] = ho;
    cn[k] = co;
  }
  __builtin_nontemporal_store(hn, &Hout[i]);
  __builtin_nontemporal_store(cn, &Cout[i]);
}

// Scalar tail for n % 4 != 0 (not hit with N*D = 51.2M, but kept for safety).
__global__ void rrgcn_gates_tail(const float* __restrict__ X,
                                 const float* __restrict__ H,
                                 const float* __restrict__ C,
                                 float* __restrict__ Hout,
                                 float* __restrict__ Cout,
                                 long long base, long long n) {
  long long i = base + threadIdx.x;
  if (i < n) {
    float ho, co;
    gate_update(X[i], H[i], C[i], ho, co);
    Hout[i] = ho;
    Cout[i] = co;
  }
}

extern "C" void kernel_launch(void* const* d_in, const int* in_sizes, int n_in,
                              void* d_out, int out_size, void* d_ws,
                              size_t ws_size, hipStream_t stream) {
  // setup_inputs order: X, edge_attr, global_attr, H, C
  // (edge_attr / global_attr provably do not affect the output.)
  const float* X = (const float*)d_in[0];
  const float* H = (const float*)d_in[3];
  const float* C = (const float*)d_in[4];

  long long n = (long long)in_sizes[0];  // N_NODES * D = 51,200,000 elements
  float* Hout = (float*)d_out;           // first n elements
  float* Cout = Hout + n;                // second n elements (16B aligned: n%4==0)

  long long nvec = n >> 2;               // 12.8M float4s
  long long rem  = n & 3LL;

  const int threads = 256;               // 8 wave32 waves per block
  long long blocks  = (nvec + threads - 1) / threads;  // 50,000 blocks
  if (blocks < 1) blocks = 1;

  if (nvec > 0) {
    rrgcn_gates_v4<<<(unsigned int)blocks, threads, 0, stream>>>(
        (const v4f*)X, (const v4f*)H, (const v4f*)C,
        (v4f*)Hout, (v4f*)Cout, (unsigned)nvec);
  }
  if (rem > 0) {
    rrgcn_gates_tail<<<1, 64, 0, stream>>>(X, H, C, Hout, Cout, n - rem, n);
  }
}